// MinGRU2_10247791968755
// MI455X (gfx1250) — compile-verified
//
#include <hip/hip_runtime.h>
#include <hip/hip_bf16.h>

// ---------------------------------------------------------------------------
// minGRU bidirectional scan, single pass, HBM-bound (~192 MiB => ~8.2 us @ 23.3 TB/s)
// x: [8, 512, 8192] f32.  Channels [0,128)=hf, [128,256)=gf, [256,384)=hb, [384,512)=gb.
// out: [8, 256, 8192] f32. [0,128)=fwd scan, [128,256)=reversed scan written back reversed.
// One wave32 per sequence (2048 waves). Async global->LDS ring buffer (CDNA5 ASYNCcnt path),
// lane-coarsened (8 elems/lane) Hillis-Steele wave scan of the (a,b) affine composition.
// ---------------------------------------------------------------------------

#define LANES 32
#define WAVES 8           // waves per block
#define CHUNK 256         // timesteps per wave per iteration
#define VPT   8           // timesteps per lane
#define NBUF  4           // LDS ring depth
#define PF    3           // prefetch distance (chunks)

constexpr int Bn  = 8;
constexpr int Hn  = 512;
constexpr int Ln  = 8192;
constexpr int NCH = Ln / CHUNK;   // 32

#if defined(__HIP_DEVICE_COMPILE__)
#if __has_builtin(__builtin_amdgcn_global_load_async_to_lds_b128) && \
    __has_builtin(__builtin_amdgcn_s_wait_asynccnt)
#define USE_ASYNC 1
#endif
#endif

typedef int v4i __attribute__((vector_size(16)));
#ifdef USE_ASYNC
typedef __attribute__((address_space(1))) v4i* gv4i_p;   // global int4*
typedef __attribute__((address_space(3))) v4i* lv4i_p;   // LDS int4*
#endif

__device__ __forceinline__ void async_copy16(const float* g, float* l) {
#ifdef USE_ASYNC
  // global_load_async_to_lds_b128 : tracked by ASYNCcnt
  __builtin_amdgcn_global_load_async_to_lds_b128((gv4i_p)g, (lv4i_p)l, 0, 0);
#else
  *(float4*)l = *(const float4*)g;   // synchronous fallback
#endif
}

template <int N>
__device__ __forceinline__ void wait_async() {
#ifdef USE_ASYNC
  __builtin_amdgcn_s_wait_asynccnt((unsigned short)N);
#endif
  __asm__ volatile("" ::: "memory");
}

__global__ __launch_bounds__(WAVES * LANES) void mingru_scan_kernel(
    const float* __restrict__ x, float* __restrict__ out) {
  __shared__ __align__(16) float smem[WAVES][NBUF][2][CHUNK];  // 64 KiB

  const int lane = threadIdx.x & (LANES - 1);
  const int wv   = threadIdx.x >> 5;
  const int seq  = blockIdx.x * WAVES + wv;       // 0..2047
  const int b    = seq >> 8;                      // /256
  const int rem  = seq & 255;
  const int dir  = rem >> 7;                      // 0 = fwd, 1 = bwd
  const int c    = rem & 127;

  const float* Hp = x + (size_t)(b * Hn + dir * 256 + c) * Ln;      // h channel
  const float* Gp = Hp + (size_t)128 * Ln;                          // g channel (+128)
  float*       Op = out + (size_t)(b * 256 + dir * 128 + c) * Ln;

  auto phys_start = [&](int ck) -> int {
    return (dir == 0) ? ck * CHUNK : (Ln - CHUNK - ck * CHUNK);
  };

  auto issue = [&](int ck) {
    const int    buf = ck & (NBUF - 1);
    const int    ps  = phys_start(ck);
    const float* gh  = Hp + ps;
    const float* gg  = Gp + ps;
    float*       lh  = &smem[wv][buf][0][0];
    float*       lg  = &smem[wv][buf][1][0];
    // 2 x b128 per array: 32 lanes * 16B = 512B each, 1 KiB per array per chunk
    async_copy16(gh + lane * 4,       lh + lane * 4);
    async_copy16(gh + 128 + lane * 4, lh + 128 + lane * 4);
    async_copy16(gg + lane * 4,       lg + lane * 4);
    async_copy16(gg + 128 + lane * 4, lg + 128 + lane * 4);
  };

  issue(0);
  issue(1);
  issue(2);

  float hcarry = 0.0f;

  for (int ck = 0; ck < NCH; ++ck) {
    if (ck + PF < NCH) {
      issue(ck + PF);
      wait_async<4 * PF>();   // <=12 outstanding => chunk ck's 4 loads landed
    } else {
      wait_async<0>();
    }

    const int     buf = ck & (NBUF - 1);
    const float4* lh4 = (const float4*)&smem[wv][buf][0][0];
    const float4* lg4 = (const float4*)&smem[wv][buf][1][0];
    const int     o   = lane * VPT;

    float h[VPT], g[VPT];
    if (dir == 0) {
      float4 h0 = lh4[o >> 2], h1 = lh4[(o >> 2) + 1];
      float4 g0 = lg4[o >> 2], g1 = lg4[(o >> 2) + 1];
      h[0] = h0.x; h[1] = h0.y; h[2] = h0.z; h[3] = h0.w;
      h[4] = h1.x; h[5] = h1.y; h[6] = h1.z; h[7] = h1.w;
      g[0] = g0.x; g[1] = g0.y; g[2] = g0.z; g[3] = g0.w;
      g[4] = g1.x; g[5] = g1.y; g[6] = g1.z; g[7] = g1.w;
    } else {
      // logical q = o+j lives at LDS index 255-o-j (chunk loaded in physical order)
      float4 h0 = lh4[(252 - o) >> 2], h1 = lh4[(248 - o) >> 2];
      float4 g0 = lg4[(252 - o) >> 2], g1 = lg4[(248 - o) >> 2];
      h[0] = h0.w; h[1] = h0.z; h[2] = h0.y; h[3] = h0.x;
      h[4] = h1.w; h[5] = h1.z; h[6] = h1.y; h[7] = h1.x;
      g[0] = g0.w; g[1] = g0.z; g[2] = g0.y; g[3] = g0.x;
      g[4] = g1.w; g[5] = g1.z; g[6] = g1.y; g[7] = g1.x;
    }

    // a = 1 - sigmoid(g), b = sigmoid(g) * h   (NaN-safe at extreme gates)
    float a[VPT], bv[VPT];
#pragma unroll
    for (int j = 0; j < VPT; ++j) {
      float e = __expf(-g[j]);                     // v_exp_f32
      float s = __builtin_amdgcn_rcpf(1.0f + e);   // sigmoid(g)
      a[j]  = 1.0f - s;
      bv[j] = s * h[j];
    }

    // lane-local composition of 8 steps: state' = a*state + b
    float A = a[0], Bv = bv[0];
#pragma unroll
    for (int j = 1; j < VPT; ++j) {
      Bv = fmaf(a[j], Bv, bv[j]);
      A *= a[j];
    }

    // inclusive wave scan of affine pairs: (A2,B2)o(A1,B1) = (A2*A1, A2*B1+B2)
#pragma unroll
    for (int d = 1; d < LANES; d <<= 1) {
      float pA = __shfl_up(A, d, LANES);
      float pB = __shfl_up(Bv, d, LANES);
      if (lane >= d) {
        Bv = fmaf(A, pB, Bv);
        A *= pA;
      }
    }
    // exclusive prefix for this lane, chunk totals from lane 31
    float eA = __shfl_up(A, 1, LANES);
    float eB = __shfl_up(Bv, 1, LANES);
    if (lane == 0) { eA = 1.0f; eB = 0.0f; }
    const float tA = __shfl(A, LANES - 1, LANES);
    const float tB = __shfl(Bv, LANES - 1, LANES);

    // replay lane's 8 elements with correct incoming state
    float hh = fmaf(eA, hcarry, eB);
    float r[VPT];
#pragma unroll
    for (int j = 0; j < VPT; ++j) {
      hh   = fmaf(a[j], hh, bv[j]);
      r[j] = hh;
    }
    hcarry = fmaf(tA, hcarry, tB);   // identical on all lanes (broadcast totals)

    if (dir == 0) {
      const int p0 = ck * CHUNK + o;
      *(float4*)(Op + p0)     = make_float4(r[0], r[1], r[2], r[3]);
      *(float4*)(Op + p0 + 4) = make_float4(r[4], r[5], r[6], r[7]);
    } else {
      // logical q -> t = L-1-q ; mirrored contiguous 16B stores
      const int tbase = Ln - 1 - (ck * CHUNK + o);
      *(float4*)(Op + tbase - 3) = make_float4(r[3], r[2], r[1], r[0]);
      *(float4*)(Op + tbase - 7) = make_float4(r[7], r[6], r[5], r[4]);
    }
  }
}

extern "C" void kernel_launch(void* const* d_in, const int* in_sizes, int n_in,
                              void* d_out, int out_size, void* d_ws, size_t ws_size,
                              hipStream_t stream) {
  (void)in_sizes; (void)n_in; (void)d_ws; (void)ws_size; (void)out_size;
  const float* x  = (const float*)d_in[0];
  float*       out = (float*)d_out;
  // 2048 sequences, 8 waves (256 threads) per block -> 256 blocks
  dim3 grid(Bn * 2 * 128 / WAVES);   // 256
  dim3 block(WAVES * LANES);         // 256
  mingru_scan_kernel<<<grid, block, 0, stream>>>(x, out);
}